// KDE_23708219474342
// MI455X (gfx1250) — compile-verified
//
#include <hip/hip_runtime.h>

typedef __attribute__((ext_vector_type(2))) float v2f;
typedef __attribute__((ext_vector_type(8))) float v8f;

#define KDE_N 4096
#define KDE_B 8
#define ROWS_PER_WAVE 2      // 16-row tiles per wave
#define WAVES_PER_BLOCK 8
#define BLOCK_SIZE (WAVES_PER_BLOCK * 32)
#define BLOCKS_PER_BATCH ((KDE_N / (16 * ROWS_PER_WAVE)) / WAVES_PER_BLOCK)  // 16

// pdf[b,i] = norm * sum_j exp(-||x_i - x_j||^2 / (2 sigma^2)) * w[b,j]
//
// Exponent tile via V_WMMA_F32_16X16X4_F32:
//   E(16x16) = A(16x4) x B(4x16),  E_ij = -d2_ij * s * log2(e)
//   A row i  = [2sE*x_i, 2sE*y_i, -sE*|x_i|^2, -sE]   (sE = s*log2e)
//   B col j  = [x_j, y_j, 1, |x_j|^2]
// kernel_ij = exp2(E_ij)  (single v_exp_f32 per element).
//
// All 8 waves of a block sweep the same batch, so the block stages the whole
// batch's per-column data [x, y, |x|^2, w] into LDS once (64 KB of the WGP's
// 320 KB). The hot loop then does one ds_load_b128 per lane per tile instead
// of two global loads, eliminating the per-iteration VMEM round-trip the
// compiler refused to pipeline.
__global__ __launch_bounds__(BLOCK_SIZE) void KDE_23708219474342_kernel(
    const float* __restrict__ weights,   // [B, N]
    const float* __restrict__ coords,    // [B, N, 2]
    const float* __restrict__ sigma_p,   // [1]
    float* __restrict__ out)             // [B, N]
{
    __shared__ float4 stile[KDE_N];      // [x, y, |x|^2, w] per column: 64 KB

    const int lane          = threadIdx.x & 31;
    const int wave_in_block = threadIdx.x >> 5;

    const int b    = blockIdx.x / BLOCKS_PER_BATCH;                       // batch
    const int t    = (blockIdx.x % BLOCKS_PER_BATCH) * WAVES_PER_BLOCK + wave_in_block;
    const int row0 = t * 16 * ROWS_PER_WAVE;

    const float sigma  = *sigma_p;
    const float inv2s2 = 1.0f / (2.0f * sigma * sigma);
    const float norm   = 1.0f / (2.0f * 3.14159265358979323846f * sigma * sigma);
    const float sE     = inv2s2 * 1.4426950408889634f;  // fold log2(e) into exponent

    const float2* __restrict__ cb = (const float2*)(coords + (size_t)b * KDE_N * 2);
    const float*  __restrict__ wb = weights + (size_t)b * KDE_N;

    // ---- Cooperative staging: whole batch's column data -> LDS ----
    for (int j = threadIdx.x; j < KDE_N; j += BLOCK_SIZE) {
        const float2 c = cb[j];
        const float  w = wb[j];
        stile[j] = make_float4(c.x, c.y, c.x * c.x + c.y * c.y, w);
    }

    const bool lo  = lane < 16;
    const int  l16 = lane & 15;

    // ---- Build A tiles (held in VGPRs for the whole column sweep) ----
    // f32 16x4 A layout: lanes 0-15 hold (K0,K1), lanes 16-31 hold (K2,K3), M = lane%16.
    v2f A[ROWS_PER_WAVE];
#pragma unroll
    for (int r = 0; r < ROWS_PER_WAVE; ++r) {
        const int m = row0 + r * 16 + l16;
        const float2 c = cb[m];
        const float sq = c.x * c.x + c.y * c.y;
        A[r].x = lo ? (2.0f * sE * c.x) : (-sE * sq);
        A[r].y = lo ? (2.0f * sE * c.y) : (-sE);
    }

    v8f acc[ROWS_PER_WAVE];
#pragma unroll
    for (int r = 0; r < ROWS_PER_WAVE; ++r) acc[r] = (v8f)0.0f;

    __syncthreads();

    // ---- Sweep all 256 column tiles out of LDS ----
    const float4* __restrict__ sp = stile + l16;
#pragma unroll 2
    for (int jt = 0; jt < KDE_N; jt += 16) {
        const float4 q = sp[jt];         // ds_load_b128: x, y, sq, w

        // f32 4x16 B layout (mirror of A): N = lane%16;
        // lanes 0-15 hold (K0,K1) = (x, y); lanes 16-31 hold (K2,K3) = (1, sq).
        v2f Bm;
        Bm.x = lo ? q.x : 1.0f;
        Bm.y = lo ? q.y : q.z;
        const float wj = q.w;

#pragma unroll
        for (int r = 0; r < ROWS_PER_WAVE; ++r) {
            v8f e = (v8f)0.0f;
            // 8 args: (neg_a, A, neg_b, B, c_mod, C, reuse_a, reuse_b)
            e = __builtin_amdgcn_wmma_f32_16x16x4_f32(
                    false, A[r], false, Bm, (short)0, e, false, false);
#pragma unroll
            for (int v = 0; v < 8; ++v) {
                // Gaussian on the trans pipe, weighted accumulate on the VALU pipe.
                acc[r][v] += __builtin_amdgcn_exp2f(e[v]) * wj;
            }
        }
    }

    // ---- Row reduction: sum acc across the 16 lanes of each half-wave ----
    // VGPR v of acc[r]: lanes 0-15 hold row (row0+16r+v), lanes 16-31 hold row (+8).
#pragma unroll
    for (int r = 0; r < ROWS_PER_WAVE; ++r) {
#pragma unroll
        for (int v = 0; v < 8; ++v) {
            float s = acc[r][v];
            s += __shfl_xor(s, 8, 32);
            s += __shfl_xor(s, 4, 32);
            s += __shfl_xor(s, 2, 32);
            s += __shfl_xor(s, 1, 32);
            if (l16 == 0) {
                const int m = row0 + r * 16 + v + (lo ? 0 : 8);
                out[(size_t)b * KDE_N + m] = norm * s;
            }
        }
    }
}

extern "C" void kernel_launch(void* const* d_in, const int* in_sizes, int n_in,
                              void* d_out, int out_size, void* d_ws, size_t ws_size,
                              hipStream_t stream) {
    (void)in_sizes; (void)n_in; (void)d_ws; (void)ws_size; (void)out_size;
    const float* weights = (const float*)d_in[0];   // [B, N]
    const float* coords  = (const float*)d_in[1];   // [B, N, 2]
    const float* sigma   = (const float*)d_in[2];   // [1]
    float* out = (float*)d_out;                     // [B, N]

    const int grid = KDE_B * BLOCKS_PER_BATCH;      // 128 blocks, 8 waves each
    KDE_23708219474342_kernel<<<grid, BLOCK_SIZE, 0, stream>>>(weights, coords, sigma, out);
}